// GAT_1537598292355
// MI455X (gfx1250) — compile-verified
//
#include <hip/hip_runtime.h>
#include <math.h>

#define NN 10000
#define EE 320000
#define HH 4
#define DD 128
#define FH (HH * DD)     // 512
#define K0 256
#define KMAX 512
#define NEG_SLOPE 0.2f

typedef __attribute__((ext_vector_type(2))) float v2f;
typedef __attribute__((ext_vector_type(8))) float v8f;

// ---------------------------------------------------------------------------
// Fill kernel (emax = -inf, denom = 0, agg = 0)
// ---------------------------------------------------------------------------
__global__ void fill_kernel(float* __restrict__ p, long n, float v) {
  long i = (long)blockIdx.x * blockDim.x + threadIdx.x;
  if (i < n) p[i] = v;
}

// ---------------------------------------------------------------------------
// FP32 WMMA GEMM: out[M,Nout] = X[M,K] @ W[K,Nout]
//
// Block = 256 threads = 8 waves. Each block owns one N-tile (16 cols) and a
// group of 8 consecutive M-tiles. The K x 16 panel of W is staged in LDS once
// per block (<= 32 KB of the WGP's 320 KB) and shared by all 8 waves, so B is
// read from global exactly once per (N-tile, M-group) instead of per wave,
// and the inner loop reads it via DS instead of 2KB-strided global dwords.
//
// Per-wave 16x16 tile via V_WMMA_F32_16X16X4_F32:
//   A (16x4 f32): lane l in half h holds (M=l, K=k+2h) in v0, (M=l,K=k+2h+1) in v1.
//   B (4x16 f32): mirror with N = l.
//   C/D: VGPR r holds (M = r + 8*half, N = l).
// ---------------------------------------------------------------------------
__global__ __launch_bounds__(256) void gemm16_wmma(const float* __restrict__ X,
                                                   const float* __restrict__ W,
                                                   float* __restrict__ out,
                                                   int M, int K, int Nout) {
  __shared__ float Bs[KMAX * 16];

  const int tid = threadIdx.x;
  const int lane = tid & 31;
  const int wv = tid >> 5;                 // wave in block: 0..7
  const int ntiles = Nout >> 4;            // N tiles (32)
  const int mg = blockIdx.x / ntiles;      // M-tile group of 8
  const int tn = blockIdx.x - mg * ntiles; // N tile

  // Cooperative stage of the K x 16 B panel into LDS.
  const float* wp = W + tn * 16;
  for (int i = tid; i < K * 16; i += 256) {
    int kk = i >> 4, c = i & 15;
    Bs[i] = wp[(size_t)kk * Nout + c];
  }
  __syncthreads();

  const int tm = mg * 8 + wv;
  if (tm * 16 >= M) return;   // partial last M-group (after the barrier)

  const int half = lane >> 4;
  const int l = lane & 15;

  const float* aptr = X + (size_t)(tm * 16 + l) * K + half * 2;
  const float* bbase = Bs + (half * 2) * 16 + l;

  v8f acc = {};
  for (int k = 0; k < K; k += 4) {
    float2 av = *(const float2*)aptr;
    v2f a; a.x = av.x; a.y = av.y;
    v2f b; b.x = bbase[0]; b.y = bbase[16];
    acc = __builtin_amdgcn_wmma_f32_16x16x4_f32(
        /*neg_a=*/false, a, /*neg_b=*/false, b,
        /*c_mod=*/(short)0, acc, /*reuse_a=*/false, /*reuse_b=*/false);
    aptr += 4;
    bbase += 64;              // 4 K-rows * 16 cols
  }

  float* obase = out + (size_t)(tm * 16 + half * 8) * Nout + tn * 16 + l;
#pragma unroll
  for (int r = 0; r < 8; ++r) obase[(size_t)r * Nout] = acc[r];
}

// ---------------------------------------------------------------------------
// el[n,h] = dot(feat[n,h,:], al[h,:]);  er likewise
// ---------------------------------------------------------------------------
__global__ void attn_coef(const float* __restrict__ feat,
                          const float* __restrict__ al,
                          const float* __restrict__ ar,
                          float* __restrict__ el, float* __restrict__ er) {
  int i = blockIdx.x * blockDim.x + threadIdx.x;  // over N*H
  if (i >= NN * HH) return;
  int h = i & (HH - 1);
  const float4* f = (const float4*)(feat + (size_t)i * DD);
  const float4* a = (const float4*)(al + h * DD);
  const float4* b = (const float4*)(ar + h * DD);
  float sl = 0.f, sr = 0.f;
#pragma unroll 4
  for (int d = 0; d < DD / 4; ++d) {
    float4 fv = f[d], av = a[d], bv = b[d];
    sl += fv.x * av.x + fv.y * av.y + fv.z * av.z + fv.w * av.w;
    sr += fv.x * bv.x + fv.y * bv.y + fv.z * bv.z + fv.w * bv.w;
  }
  el[i] = sl;
  er[i] = sr;
}

// ---------------------------------------------------------------------------
// float atomic max via monotone int trick (emax initialized to -inf)
// ---------------------------------------------------------------------------
__device__ __forceinline__ void atomicMaxF(float* addr, float val) {
  if (val >= 0.f)
    atomicMax((int*)addr, __float_as_int(val));
  else
    atomicMin((unsigned int*)addr, __float_as_uint(val));
}

__device__ __forceinline__ float leaky(float v) {
  return v > 0.f ? v : NEG_SLOPE * v;
}

__global__ void edge_max(const int* __restrict__ src, const int* __restrict__ dst,
                         const float* __restrict__ el, const float* __restrict__ er,
                         float* __restrict__ emax) {
  int e = blockIdx.x * blockDim.x + threadIdx.x;
  if (e >= EE) return;
  int s = src[e], t = dst[e];
#pragma unroll
  for (int h = 0; h < HH; ++h) {
    float v = leaky(el[s * HH + h] + er[t * HH + h]);
    atomicMaxF(&emax[t * HH + h], v);
  }
}

__global__ void edge_exp(const int* __restrict__ src, const int* __restrict__ dst,
                         const float* __restrict__ el, const float* __restrict__ er,
                         const float* __restrict__ emax,
                         float* __restrict__ exv, float* __restrict__ denom) {
  int e = blockIdx.x * blockDim.x + threadIdx.x;
  if (e >= EE) return;
  int s = src[e], t = dst[e];
#pragma unroll
  for (int h = 0; h < HH; ++h) {
    float v = leaky(el[s * HH + h] + er[t * HH + h]);
    float m = emax[t * HH + h];           // finite for any dst that has edges
    float xo = __expf(v - m);
    exv[(size_t)e * HH + h] = xo;
    atomicAdd(&denom[t * HH + h], xo);
  }
}

// One block (128 threads = one D slice) per edge; heads unrolled.
__global__ __launch_bounds__(128) void edge_scatter(
    const int* __restrict__ src, const int* __restrict__ dst,
    const float* __restrict__ feat, const float* __restrict__ exv,
    const float* __restrict__ denom, float* __restrict__ agg) {
  int e = blockIdx.x;
  int d = threadIdx.x;
  int s = src[e], t = dst[e];
  const float* fs = feat + (size_t)s * FH + d;
  float* ob = agg + (size_t)t * FH + d;
  __builtin_prefetch(fs, 0, 0);
#pragma unroll
  for (int h = 0; h < HH; ++h) {
    float dn = denom[t * HH + h];
    float alpha = exv[(size_t)e * HH + h] / (dn == 0.f ? 1.f : dn);
    atomicAdd(ob + h * DD, fs[h * DD] * alpha);
  }
}

// ---------------------------------------------------------------------------
// ELU and finalize (residual + ELU + head-mean)
// ---------------------------------------------------------------------------
__global__ void layer_elu(const float* __restrict__ agg, float* __restrict__ out,
                          long n) {
  long i = (long)blockIdx.x * blockDim.x + threadIdx.x;
  if (i < n) {
    float v = agg[i];
    out[i] = v > 0.f ? v : (__expf(v) - 1.f);
  }
}

__global__ void finalize(const float* __restrict__ agg,
                         const float* __restrict__ hin,
                         float* __restrict__ out) {
  int i = blockIdx.x * blockDim.x + threadIdx.x;  // over N*D
  if (i >= NN * DD) return;
  int n = i / DD, d = i - n * DD;
  float s = 0.f;
#pragma unroll
  for (int h = 0; h < HH; ++h) {
    size_t off = (size_t)n * FH + h * DD + d;
    float v = agg[off] + hin[off];
    s += v > 0.f ? v : (__expf(v) - 1.f);
  }
  out[i] = s * (1.f / HH);
}

// ---------------------------------------------------------------------------
// Launch
// ---------------------------------------------------------------------------
extern "C" void kernel_launch(void* const* d_in, const int* in_sizes, int n_in,
                              void* d_out, int out_size, void* d_ws, size_t ws_size,
                              hipStream_t stream) {
  const int* src = (const int*)d_in[0];
  const int* dst = (const int*)d_in[1];
  const float* x   = (const float*)d_in[2];
  const float* W0  = (const float*)d_in[3];
  const float* al0 = (const float*)d_in[4];
  const float* ar0 = (const float*)d_in[5];
  const float* W1  = (const float*)d_in[6];
  const float* al1 = (const float*)d_in[7];
  const float* ar1 = (const float*)d_in[8];
  float* out = (float*)d_out;

  float* ws    = (float*)d_ws;
  float* feat  = ws;                        // N*512
  float* agg   = feat  + (size_t)NN * FH;   // N*512
  float* hbuf  = agg   + (size_t)NN * FH;   // N*512
  float* el    = hbuf  + (size_t)NN * FH;   // N*H
  float* er    = el    + (size_t)NN * HH;
  float* emax  = er    + (size_t)NN * HH;
  float* denom = emax  + (size_t)NN * HH;
  float* exv   = denom + (size_t)NN * HH;   // E*H

  const int mtiles = (NN + 15) / 16;        // 625
  const int mgroups = (mtiles + 7) / 8;     // 79
  const int ntiles = FH / 16;               // 32
  const int gemmBlocks = mgroups * ntiles;  // 2528

  auto run_layer = [&](const float* xin, const float* W, int K,
                       const float* al, const float* ar) {
    gemm16_wmma<<<gemmBlocks, 256, 0, stream>>>(xin, W, feat, NN, K, FH);
    attn_coef<<<(NN * HH + 255) / 256, 256, 0, stream>>>(feat, al, ar, el, er);
    fill_kernel<<<(NN * HH + 255) / 256, 256, 0, stream>>>(emax, (long)NN * HH,
                                                           -__builtin_huge_valf());
    fill_kernel<<<(NN * HH + 255) / 256, 256, 0, stream>>>(denom, (long)NN * HH, 0.f);
    fill_kernel<<<((long)NN * FH + 255) / 256, 256, 0, stream>>>(agg, (long)NN * FH, 0.f);
    edge_max<<<(EE + 255) / 256, 256, 0, stream>>>(src, dst, el, er, emax);
    edge_exp<<<(EE + 255) / 256, 256, 0, stream>>>(src, dst, el, er, emax, exv, denom);
    edge_scatter<<<EE, 128, 0, stream>>>(src, dst, feat, exv, denom, agg);
  };

  // Layer 0
  run_layer(x, W0, K0, al0, ar0);
  layer_elu<<<((long)NN * FH + 255) / 256, 256, 0, stream>>>(agg, hbuf, (long)NN * FH);
  // Layer 1 (+ residual & head-mean in finalize)
  run_layer(hbuf, W1, FH, al1, ar1);
  finalize<<<(NN * DD + 255) / 256, 256, 0, stream>>>(agg, hbuf, out);
}